// expand_bitPackets_29841432772984
// MI455X (gfx1250) — compile-verified
//
#include <hip/hip_runtime.h>
#include <hip/hip_bf16.h>

// Problem constants (from reference)
#define BATCH   4
#define XL      1514
#define KH      50
#define KWID    4
#define C_OUT   20
#define HOUT    1451   // 1500 - 49
#define WOUT    5      // 8 - 3
#define KDIM    145100 // HOUT*WOUT*C_OUT
#define NOUT    1500

// GEMM tiling
#define NTILE   94     // ceil(1500/16)
#define KCHUNK  16384
#define NCHUNK  9      // ceil(145100/16384)
#define KSTAGE  448    // k-values staged in LDS per iteration (8 waves x 56)
#define KSLICE  56     // k-values per wave per stage (14 WMMAs)
#define WSTR    452    // padded LDS row stride: 452%64==4 -> conflict-free frag reads
#define N4      (KSTAGE / 4)

typedef __attribute__((ext_vector_type(2))) float v2f;
typedef __attribute__((ext_vector_type(4))) float v4f;
typedef __attribute__((ext_vector_type(8))) float v8f;

// ---------------------------------------------------------------------------
// Kernel 1: bit-unpack + conv2d(1->20, 50x4, valid) -> flat[b*KDIM + c*7255 + h*5 + w]
// ---------------------------------------------------------------------------
__global__ void __launch_bounds__(256)
conv_bits_kernel(const float* __restrict__ x, const float* __restrict__ cw,
                 const float* __restrict__ cb, float* __restrict__ flat) {
    int idx = blockIdx.x * 256 + threadIdx.x;
    if (idx >= BATCH * KDIM) return;
    int b  = idx / KDIM;
    int r  = idx - b * KDIM;
    int c  = r / (HOUT * WOUT);
    int r2 = r - c * (HOUT * WOUT);
    int h  = r2 / WOUT;
    int w  = r2 - h * WOUT;

    const float* xr = x + b * XL;
    const float* wk = cw + c * (KH * KWID);
    float acc = cb[c];
    for (int kh = 0; kh < KH; ++kh) {
        int byte = (int)xr[h + kh];
#pragma unroll
        for (int kw = 0; kw < KWID; ++kw) {
            float bit = (float)((byte >> (7 - (w + kw))) & 1);
            acc += bit * wk[kh * KWID + kw];
        }
    }
    flat[idx] = acc;
}

// ---------------------------------------------------------------------------
// Kernel 2: partial GEMM  D[16x16] += A[16xK] * W^T[Kx16]  via V_WMMA_F32_16X16X4_F32
// A rows 0..3 = flat activations; rows 4..15 kept zero in LDS so the fragment
// build is a single unconditional ds_load_b64 (no EXEC masking around WMMA).
// One workgroup = (output tile, k-chunk); 8 waves split each LDS stage.
// ---------------------------------------------------------------------------
__global__ void __launch_bounds__(256)
gemm_partial_kernel(const float* __restrict__ flat, const float* __restrict__ fcw,
                    float* __restrict__ partial) {
    __shared__ float w_lds[16 * WSTR];   // fc_w tile: 16 rows x KSTAGE (padded)
    __shared__ float a_lds[16 * WSTR];   // activation tile; rows 4..15 stay zero
    float* red = a_lds + 4 * WSTR;       // reduction buffer aliased into zero rows

    const int tid   = threadIdx.x;
    const int lane  = tid & 31;
    const int wids  = __builtin_amdgcn_readfirstlane(tid >> 5);
    const int tile  = __builtin_amdgcn_readfirstlane(blockIdx.x / NCHUNK);
    const int chunk = __builtin_amdgcn_readfirstlane(blockIdx.x - tile * NCHUNK);
    const int obase = tile * 16;
    const int kbeg  = chunk * KCHUNK;
    int kend = kbeg + KCHUNK; if (kend > KDIM) kend = KDIM;

    // WMMA 16x16x4 f32 fragment lane mapping:
    //   A: lane = M + 16*khi holds K = {2*khi, 2*khi+1}
    //   B: lane = N + 16*khi holds K = {2*khi, 2*khi+1}
    const int m  = lane & 15;
    const int kh = (lane >> 4) << 1;
    const float* ap = &a_lds[m * WSTR + kh];
    const float* wp = &w_lds[m * WSTR + kh];

    // Zero the padding rows of the A tile once (rows 4..15).
    for (int i = tid; i < 12 * WSTR; i += 256)
        a_lds[4 * WSTR + i] = 0.f;

    v8f acc = {};

    for (int k0 = kbeg; k0 < kend; k0 += KSTAGE) {
        int slen = kend - k0; if (slen > KSTAGE) slen = KSTAGE;
        slen = __builtin_amdgcn_readfirstlane(slen);
        __syncthreads();

        // Stage fc_w rows: 16 x 112 float4 slots, coalesced; zero-pad OOB.
#pragma unroll
        for (int j = 0; j < (16 * N4) / 256; ++j) {
            int i   = tid + j * 256;
            int row = i / N4;
            int c4  = (i - row * N4) * 4;
            v4f val = {0.f, 0.f, 0.f, 0.f};
            int o = obase + row;
            if (c4 < slen && o < NOUT)
                val = *(const v4f*)(fcw + (size_t)o * KDIM + k0 + c4);
            *(v4f*)&w_lds[row * WSTR + c4] = val;
        }
        // Stage activations: 4 x 112 float4 slots.
        {
            int i = tid;
            if (i < 4 * N4) {
                int row = i / N4;
                int c4  = (i - row * N4) * 4;
                v4f val = {0.f, 0.f, 0.f, 0.f};
                if (c4 < slen)
                    val = *(const v4f*)(flat + (size_t)row * KDIM + k0 + c4);
                *(v4f*)&a_lds[row * WSTR + c4] = val;
            }
        }
        // Prefetch next stage of fc_w into cache (global_prefetch_b8).
        if (k0 + KSTAGE < kend) {
            int prow = tid >> 4;            // 0..15
            int pcol = (tid & 15) * 28;     // spread across the 448-float stage
            int o = obase + prow; if (o >= NOUT) o = NOUT - 1;
            __builtin_prefetch(fcw + (size_t)o * KDIM + k0 + KSTAGE + pcol, 0, 1);
        }
        __syncthreads();

        // Each wave covers k-slice [wids*56, wids*56+56): 14 WMMAs, unrolled.
        const int ks = wids * KSLICE;
        int ke = ks + KSLICE; if (ke > slen) ke = slen;
        if (ke - ks == KSLICE) {
#pragma unroll
            for (int u = 0; u < KSLICE / 4; ++u) {
                int kk = ks + u * 4;
                v2f a = *(const v2f*)(ap + kk);
                v2f b = *(const v2f*)(wp + kk);
                acc = __builtin_amdgcn_wmma_f32_16x16x4_f32(
                    false, a, false, b, (short)0, acc, false, false);
            }
        } else {
            for (int kk = ks; kk < ke; kk += 4) {
                v2f a = *(const v2f*)(ap + kk);
                v2f b = *(const v2f*)(wp + kk);
                acc = __builtin_amdgcn_wmma_f32_16x16x4_f32(
                    false, a, false, b, (short)0, acc, false, false);
            }
        }
    }

    // Deterministic cross-wave reduction (red aliases A rows 4..15, done with them).
    __syncthreads();
#pragma unroll
    for (int v = 0; v < 8; ++v)
        red[wids * 256 + v * 32 + lane] = acc[v];
    __syncthreads();
    float s = 0.f;
#pragma unroll
    for (int w = 0; w < 8; ++w)
        s += red[w * 256 + tid];
    partial[((size_t)(tile * NCHUNK + chunk) << 8) + tid] = s;
}

// ---------------------------------------------------------------------------
// Kernel 3: reduce chunk partials, add bias, splice into output [4,1,1514].
// D element (M=b, N=n) lives at j = v*32 + lane with v=b (b<8), lane=n (n<16).
// ---------------------------------------------------------------------------
__global__ void __launch_bounds__(256)
finalize_kernel(const float* __restrict__ x, const float* __restrict__ fcb,
                const float* __restrict__ partial, float* __restrict__ out) {
    int o = blockIdx.x * 256 + threadIdx.x;
    if (o >= XL) return;
    if (o < NOUT) {
        int t = o >> 4;
        int n = o & 15;
        float bias = fcb[o];
#pragma unroll
        for (int b = 0; b < BATCH; ++b) {
            float s = bias;
            for (int c = 0; c < NCHUNK; ++c)
                s += partial[((size_t)(t * NCHUNK + c) << 8) + b * 32 + n];
            out[b * XL + o] = s;
        }
    } else {
#pragma unroll
        for (int b = 0; b < BATCH; ++b)
            out[b * XL + o] = x[b * XL + o];
    }
}

// ---------------------------------------------------------------------------
extern "C" void kernel_launch(void* const* d_in, const int* in_sizes, int n_in,
                              void* d_out, int out_size, void* d_ws, size_t ws_size,
                              hipStream_t stream) {
    const float* x      = (const float*)d_in[0]; // [4,1,1514]
    const float* conv_w = (const float*)d_in[1]; // [20,1,50,4]
    const float* conv_b = (const float*)d_in[2]; // [20]
    const float* fc_w   = (const float*)d_in[3]; // [1500,145100]
    const float* fc_b   = (const float*)d_in[4]; // [1500]
    float* out = (float*)d_out;                  // [4,1,1514]

    float* flat    = (float*)d_ws;                // 4*145100 floats (2.32 MB)
    float* partial = flat + (size_t)BATCH * KDIM; // 94*9*256 floats (0.87 MB)

    conv_bits_kernel<<<(BATCH * KDIM + 255) / 256, 256, 0, stream>>>(
        x, conv_w, conv_b, flat);

    gemm_partial_kernel<<<NTILE * NCHUNK, 256, 0, stream>>>(
        flat, fc_w, partial);

    finalize_kernel<<<(XL + 255) / 256, 256, 0, stream>>>(
        x, fc_b, partial, out);
}